// GCNModel_81724637708715
// MI455X (gfx1250) — compile-verified
//
#include <hip/hip_runtime.h>
#include <hip/hip_bf16.h>

typedef __attribute__((ext_vector_type(2))) float v2f;
typedef __attribute__((ext_vector_type(8))) float v8f;

#define NNODES 50000
#define NEDGES 600000
#define NGRAPH 8
#define DIM    128
#define NOUT   10
#define NEG_SLOPE 0.01f

// ---------------------------------------------------------------------------
// Generic zero fill
// ---------------------------------------------------------------------------
__global__ void zero_kernel(float* __restrict__ p, int n) {
    int i = blockIdx.x * blockDim.x + threadIdx.x;
    if (i < n) p[i] = 0.0f;
}

// ---------------------------------------------------------------------------
// Degree accumulation (float counts; exact up to 2^24)
// ---------------------------------------------------------------------------
__global__ void degree_kernel(const int* __restrict__ src, const int* __restrict__ dst,
                              float* __restrict__ sdeg, float* __restrict__ ddeg, int nE) {
    int e = blockIdx.x * blockDim.x + threadIdx.x;
    if (e >= nE) return;
    unsafeAtomicAdd(&sdeg[src[e]], 1.0f);
    unsafeAtomicAdd(&ddeg[dst[e]], 1.0f);
}

// deg -> clip(deg,1)^-0.5, in place over a contiguous span
__global__ void norm_kernel(float* __restrict__ p, int n) {
    int i = blockIdx.x * blockDim.x + threadIdx.x;
    if (i < n) p[i] = rsqrtf(fmaxf(p[i], 1.0f));
}

// ---------------------------------------------------------------------------
// Y[row, :] = (srcn[row] * X[row, :]) @ W   using V_WMMA_F32_16X16X4_F32
//
// LDS holds W in K-pair-interleaved form:
//   Wp[kp*256 + 2n + j] = W[2*kp + j][n]   (kp = 0..63, n = 0..127, j = 0..1)
// so a WMMA B fragment ({W[k][n], W[k+1][n]} in an adjacent even-aligned VGPR
// pair) is exactly one ds_load_b64 — no repacking movs.
//
// 8 waves/block; wave w computes rows [blk*128 + w*16, +16) x all 128 cols.
// A frag (16x4 fp32): lane m = lane&15 row, kb = (lane>>4)*2 K-pair  -> v2f
// C/D (16x16 fp32):   VGPR r holds M=r (lanes 0-15) / M=r+8 (lanes 16-31)
// A loads are software-prefetched 2 steps deep to hide global latency behind
// the 8 WMMAs + 8 DS loads of each K-step.
// ---------------------------------------------------------------------------
__global__ __launch_bounds__(256) void gemm_norm_kernel(
        const float* __restrict__ X,     // [nrows, 128]
        const float* __restrict__ W,     // [128, 128]
        const float* __restrict__ srcn,  // [nrows]
        float* __restrict__ Y,           // [nrows, 128]
        int nrows) {
    __shared__ float Wp[64 * 256];  // 64 KB (WGP budget is 320 KB)

    // cooperative interleaved load: 2048 chunks of (2 rows x 4 cols) / 256 thr
    {
        int t = threadIdx.x;
#pragma unroll
        for (int i = 0; i < 8; ++i) {
            int idx = t + i * 256;          // 0..2047
            int kp  = idx >> 5;             // 0..63
            int n4  = (idx & 31) << 2;      // 0,4,...,124
            float4 r0 = *(const float4*)(W + (size_t)(2 * kp)     * DIM + n4);
            float4 r1 = *(const float4*)(W + (size_t)(2 * kp + 1) * DIM + n4);
            float4* d = (float4*)(Wp + kp * 256 + n4 * 2);
            d[0] = make_float4(r0.x, r1.x, r0.y, r1.y);
            d[1] = make_float4(r0.z, r1.z, r0.w, r1.w);
        }
    }
    __syncthreads();

    const int wave  = threadIdx.x >> 5;
    const int lane  = threadIdx.x & 31;
    const int mtile = blockIdx.x * 8 + wave;
    const int row0  = mtile * 16;
    if (row0 >= nrows) return;  // uniform across the wave: EXEC stays all-1s

    const int m  = lane & 15;         // M row (A) / N col (B, C)
    const int kb = (lane >> 4) << 1;  // 0 or 2: which K pair this half-wave holds
    const int row = row0 + m;
    const float s = srcn[row];

    v8f acc[8] = {};  // 8 col-tiles x (16x16 f32 / 32 lanes) = 64 VGPRs

    const float* xr = X + (size_t)row * DIM + kb;
    v2f a0 = *(const v2f*)(xr);
    v2f a1 = *(const v2f*)(xr + 4);

#pragma unroll
    for (int k = 0; k < DIM; k += 4) {
        v2f a = a0;
        a0 = a1;
        if (k + 8 < DIM) a1 = *(const v2f*)(xr + k + 8);
        a.x *= s;
        a.y *= s;
        const float* brow = Wp + ((k >> 1) + (kb >> 1)) * 256 + m * 2;
#pragma unroll
        for (int ct = 0; ct < 8; ++ct) {
            v2f b = *(const v2f*)(brow + ct * 32);  // one ds_load_b64
            acc[ct] = __builtin_amdgcn_wmma_f32_16x16x4_f32(
                false, a, false, b, (short)0, acc[ct], false, false);
        }
    }

    const int rsel = (lane >> 4) * 8;  // lanes 16-31 hold rows M=8..15
#pragma unroll
    for (int ct = 0; ct < 8; ++ct) {
#pragma unroll
        for (int r = 0; r < 8; ++r) {
            Y[(size_t)(row0 + r + rsel) * DIM + ct * 16 + m] = acc[ct][r];
        }
    }
}

// ---------------------------------------------------------------------------
// SpMM scatter: B[dst[e], :] += A[src[e], :]. One wave per edge, float4/lane.
// ---------------------------------------------------------------------------
__global__ void spmm_kernel(const float* __restrict__ A, const int* __restrict__ src,
                            const int* __restrict__ dst, float* __restrict__ B, int nE) {
    int tid = blockIdx.x * blockDim.x + threadIdx.x;
    int e = tid >> 5;
    int c = tid & 31;
    if (e >= nE) return;
    int s = src[e], t = dst[e];
    const float4 v = *(const float4*)(A + (size_t)s * DIM + c * 4);
    float* out = B + (size_t)t * DIM + c * 4;
    unsafeAtomicAdd(out + 0, v.x);
    unsafeAtomicAdd(out + 1, v.y);
    unsafeAtomicAdd(out + 2, v.z);
    unsafeAtomicAdd(out + 3, v.w);
}

// B[n, d] = B[n, d] * dstn[n] + bias[d]
__global__ void finalize_kernel(float* __restrict__ B, const float* __restrict__ dstn,
                                const float* __restrict__ bias, int nN) {
    int tid = blockIdx.x * blockDim.x + threadIdx.x;
    int n = tid >> 5;
    int c = tid & 31;
    if (n >= nN) return;
    float s = dstn[n];
    float4 v = *(const float4*)(B + (size_t)n * DIM + c * 4);
    const float4 bb = *(const float4*)(bias + c * 4);
    v.x = v.x * s + bb.x;
    v.y = v.y * s + bb.y;
    v.z = v.z * s + bb.z;
    v.w = v.w * s + bb.w;
    *(float4*)(B + (size_t)n * DIM + c * 4) = v;
}

// ---------------------------------------------------------------------------
// Per-graph pooling
// ---------------------------------------------------------------------------
__global__ void pool_accum_kernel(const float* __restrict__ h, const int* __restrict__ gid,
                                  float* __restrict__ pooled, float* __restrict__ cnt, int nN) {
    int tid = blockIdx.x * blockDim.x + threadIdx.x;
    int n = tid >> 5;
    int c = tid & 31;
    if (n >= nN) return;
    int g = gid[n];
    const float4 v = *(const float4*)(h + (size_t)n * DIM + c * 4);
    float* p = pooled + g * DIM + c * 4;
    unsafeAtomicAdd(p + 0, v.x);
    unsafeAtomicAdd(p + 1, v.y);
    unsafeAtomicAdd(p + 2, v.z);
    unsafeAtomicAdd(p + 3, v.w);
    if (c == 0) unsafeAtomicAdd(&cnt[g], 1.0f);
}

__global__ void pool_div_kernel(float* __restrict__ pooled, const float* __restrict__ cnt) {
    int i = blockIdx.x * blockDim.x + threadIdx.x;
    if (i >= NGRAPH * DIM) return;
    pooled[i] /= fmaxf(cnt[i / DIM], 1.0f);
}

// ---------------------------------------------------------------------------
// Dense head: Y[m,n] = lrelu(sum_k X[m,k] * W[k,n] + b[n]); tiny (M=8)
// ---------------------------------------------------------------------------
__global__ void fc_lrelu_kernel(const float* __restrict__ X, const float* __restrict__ W,
                                const float* __restrict__ bias, float* __restrict__ Y,
                                int M, int K, int Nout) {
    int idx = blockIdx.x * blockDim.x + threadIdx.x;
    if (idx >= M * Nout) return;
    int mm = idx / Nout, nn = idx % Nout;
    float acc = bias[nn];
    for (int k = 0; k < K; ++k) acc += X[mm * K + k] * W[k * Nout + nn];
    Y[idx] = acc > 0.0f ? acc : acc * NEG_SLOPE;
}

// ---------------------------------------------------------------------------
extern "C" void kernel_launch(void* const* d_in, const int* in_sizes, int n_in,
                              void* d_out, int out_size, void* d_ws, size_t ws_size,
                              hipStream_t stream) {
    const float* inputs = (const float*)d_in[0];
    const int*   src    = (const int*)d_in[1];
    const int*   dst    = (const int*)d_in[2];
    const int*   gid    = (const int*)d_in[3];
    const float* W1  = (const float*)d_in[4];
    const float* b1  = (const float*)d_in[5];
    const float* W2  = (const float*)d_in[6];
    const float* b2  = (const float*)d_in[7];
    const float* W3  = (const float*)d_in[8];
    const float* b3  = (const float*)d_in[9];
    const float* Wc1 = (const float*)d_in[10];
    const float* bc1 = (const float*)d_in[11];
    const float* Wc2 = (const float*)d_in[12];
    const float* bc2 = (const float*)d_in[13];
    const float* Wc3 = (const float*)d_in[14];
    const float* bc3 = (const float*)d_in[15];
    float* out = (float*)d_out;

    float* ws     = (float*)d_ws;
    float* buf0   = ws;                        // [N, D]
    float* buf1   = ws + (size_t)NNODES * DIM; // [N, D]
    float* snorm  = ws + (size_t)2 * NNODES * DIM;  // [N]
    float* dnorm  = snorm + NNODES;                 // [N]  (contiguous with snorm)
    float* pooled = dnorm + NNODES;                 // [G, D]
    float* cnt    = pooled + NGRAPH * DIM;          // [G]  (contiguous with pooled)
    float* c1     = cnt + NGRAPH;                   // [G, 512]
    float* c2     = c1 + NGRAPH * 512;              // [G, 256]

    const int TB = 256;
    const int mtiles = NNODES / 16;                 // 3125 (exact)
    const int gemm_blocks = (mtiles + 7) / 8;       // 391
    const int rowchunk_threads = NNODES * 32;       // node-major float4 kernels
    const int edgechunk_threads = NEDGES * 32;

    // degrees -> norms (zero the 2N accumulator span first)
    zero_kernel<<<(2 * NNODES + TB - 1) / TB, TB, 0, stream>>>(snorm, 2 * NNODES);
    degree_kernel<<<(NEDGES + TB - 1) / TB, TB, 0, stream>>>(src, dst, snorm, dnorm, NEDGES);
    norm_kernel<<<(2 * NNODES + TB - 1) / TB, TB, 0, stream>>>(snorm, 2 * NNODES);

    // ---- layer 1 ----
    gemm_norm_kernel<<<gemm_blocks, TB, 0, stream>>>(inputs, W1, snorm, buf0, NNODES);
    zero_kernel<<<(NNODES * DIM + TB - 1) / TB, TB, 0, stream>>>(buf1, NNODES * DIM);
    spmm_kernel<<<(edgechunk_threads + TB - 1) / TB, TB, 0, stream>>>(buf0, src, dst, buf1, NEDGES);
    finalize_kernel<<<(rowchunk_threads + TB - 1) / TB, TB, 0, stream>>>(buf1, dnorm, b1, NNODES);

    // ---- layer 2 ----
    gemm_norm_kernel<<<gemm_blocks, TB, 0, stream>>>(buf1, W2, snorm, buf0, NNODES);
    zero_kernel<<<(NNODES * DIM + TB - 1) / TB, TB, 0, stream>>>(buf1, NNODES * DIM);
    spmm_kernel<<<(edgechunk_threads + TB - 1) / TB, TB, 0, stream>>>(buf0, src, dst, buf1, NEDGES);
    finalize_kernel<<<(rowchunk_threads + TB - 1) / TB, TB, 0, stream>>>(buf1, dnorm, b2, NNODES);

    // ---- layer 3 ----
    gemm_norm_kernel<<<gemm_blocks, TB, 0, stream>>>(buf1, W3, snorm, buf0, NNODES);
    zero_kernel<<<(NNODES * DIM + TB - 1) / TB, TB, 0, stream>>>(buf1, NNODES * DIM);
    spmm_kernel<<<(edgechunk_threads + TB - 1) / TB, TB, 0, stream>>>(buf0, src, dst, buf1, NEDGES);
    finalize_kernel<<<(rowchunk_threads + TB - 1) / TB, TB, 0, stream>>>(buf1, dnorm, b3, NNODES);

    // ---- mean_nodes pooling ----
    zero_kernel<<<(NGRAPH * DIM + NGRAPH + TB - 1) / TB, TB, 0, stream>>>(pooled, NGRAPH * DIM + NGRAPH);
    pool_accum_kernel<<<(rowchunk_threads + TB - 1) / TB, TB, 0, stream>>>(buf1, gid, pooled, cnt, NNODES);
    pool_div_kernel<<<(NGRAPH * DIM + TB - 1) / TB, TB, 0, stream>>>(pooled, cnt);

    // ---- classifier head (8 rows; negligible) ----
    fc_lrelu_kernel<<<(NGRAPH * 512 + TB - 1) / TB, TB, 0, stream>>>(pooled, Wc1, bc1, c1, NGRAPH, DIM, 512);
    fc_lrelu_kernel<<<(NGRAPH * 256 + TB - 1) / TB, TB, 0, stream>>>(c1, Wc2, bc2, c2, NGRAPH, 512, 256);
    fc_lrelu_kernel<<<(NGRAPH * NOUT + TB - 1) / TB, TB, 0, stream>>>(c2, Wc3, bc3, out, NGRAPH, 256, NOUT);
}